// SwinGATNet_80513456931512
// MI455X (gfx1250) — compile-verified
//
#include <hip/hip_runtime.h>
#include <math.h>

typedef __attribute__((ext_vector_type(2))) float v2f;
typedef __attribute__((ext_vector_type(8))) float v8f;

#define N_NODES   100000
#define N_EDGES   1600000
#define IN_DIM    128
#define HID       64
#define HEADS     4
#define DHEAD     16
#define NEG_SLOPE 0.2f
#define GAT_EPS   1e-9f

// ---------------------------------------------------------------------------
// C[M x 64] = act(A[M x K] @ W[K x 64] (+ bias)) via V_WMMA_F32_16X16X4_F32.
// Block = 128 threads = 4 wave32s; wave w owns output cols [16w, 16w+16)
// (one GAT head per wave), block b owns rows [16b, 16b+16). 100000 % 16 == 0
// so every wave runs with full EXEC (WMMA requirement).
//
// Fragment layouts (ISA 7.12.2):
//   A 16x4 : lanes 0-15 -> M=lane, {v0,v1}={K,K+1}; lanes 16-31 -> {K+2,K+3}
//   B 4x16 : v0 = rows {K, K+2}, v1 = rows {K+1, K+3}, N = lane&15
//   C/D    : acc[r] = row (16b + r + 8*(lane>>4)), col (16w + (lane&15))
// ---------------------------------------------------------------------------
template<int K, bool BIAS_RELU>
__global__ __launch_bounds__(128)
void gemm_wmma_f32(const float* __restrict__ A, const float* __restrict__ W,
                   const float* __restrict__ bias, float* __restrict__ C) {
  const int lane = threadIdx.x & 31;
  const int wave = threadIdx.x >> 5;
  const int row0 = blockIdx.x * 16;
  const int col0 = wave * 16;
  const int half = lane >> 4;     // 0: K,K+1   1: K+2,K+3
  const int lr   = lane & 15;

  const float* __restrict__ arow = A + (size_t)(row0 + lr) * K;

  v8f acc = {};
#pragma unroll
  for (int k = 0; k < K; k += 4) {
    const int ka = k + 2 * half;
    v2f av, bv;
    av[0] = arow[ka];
    av[1] = arow[ka + 1];
    bv[0] = W[(size_t)ka       * HID + col0 + lr];
    bv[1] = W[(size_t)(ka + 1) * HID + col0 + lr];
    // 8 args: (neg_a, A, neg_b, B, c_mod, C, reuse_a, reuse_b)
    acc = __builtin_amdgcn_wmma_f32_16x16x4_f32(
        false, av, false, bv, (short)0, acc, false, false);
  }

  const float b = BIAS_RELU ? bias[col0 + lr] : 0.0f;
#pragma unroll
  for (int r = 0; r < 8; ++r) {
    float v = acc[r] + b;
    if (BIAS_RELU) v = fmaxf(v, 0.0f);
    C[(size_t)(row0 + r + 8 * half) * HID + col0 + lr] = v;
  }
}

// Per-(node, head) attention scalars: s_src = <xp[n,h,:], att_src[h,:]>,
// s_dst likewise. att vectors (64 floats) stay resident in cache.
__global__ void node_scores(const float* __restrict__ xp,
                            const float* __restrict__ att_src,
                            const float* __restrict__ att_dst,
                            float* __restrict__ s_src,
                            float* __restrict__ s_dst) {
  const int gid = blockIdx.x * blockDim.x + threadIdx.x;
  if (gid >= N_NODES * HEADS) return;
  const int n = gid >> 2, hd = gid & 3;
  const float* __restrict__ v  = xp + (size_t)n * HID + hd * DHEAD;
  const float* __restrict__ as = att_src + hd * DHEAD;
  const float* __restrict__ ad = att_dst + hd * DHEAD;
  float ss = 0.f, sd = 0.f;
#pragma unroll
  for (int d = 0; d < DHEAD; ++d) {
    const float x = v[d];
    ss += x * as[d];
    sd += x * ad[d];
  }
  s_src[gid] = ss;
  s_dst[gid] = sd;
}

__global__ void zero_f32(float* __restrict__ p, int n) {
  const int gid = blockIdx.x * blockDim.x + threadIdx.x;
  if (gid < n) p[gid] = 0.0f;
}

__device__ __forceinline__ float edge_logit_exp(const float* s_src,
                                                const float* s_dst,
                                                int s, int d, int hd) {
  float ev = s_src[s * HEADS + hd] + s_dst[d * HEADS + hd];
  ev = ev > 0.0f ? ev : NEG_SLOPE * ev;      // LeakyReLU(0.2)
  return expf(ev);                           // no max-subtract, matches ref
}

// Pass 1: softmax denominators. One thread per (edge, head); s_* arrays are
// 1.6 MB each -> L2 (often WGP$) resident, so gathers are cheap.
__global__ void edge_pass1(const int* __restrict__ src,
                           const int* __restrict__ dst,
                           const float* __restrict__ s_src,
                           const float* __restrict__ s_dst,
                           float* __restrict__ denom) {
  const int gid = blockIdx.x * blockDim.x + threadIdx.x;
  if (gid >= N_EDGES * HEADS) return;
  const int e = gid >> 2, hd = gid & 3;
  const int s = src[e], d = dst[e];
  atomicAdd(&denom[d * HEADS + hd], edge_logit_exp(s_src, s_dst, s, d, hd));
}

// Pass 2: recompute exp(e) (cheaper than a 25.6 MB spill), normalize, and
// scatter alpha * x_j into out[dst]. xp (25.6 MB) is L2-resident on MI455X.
__global__ void edge_pass2(const int* __restrict__ src,
                           const int* __restrict__ dst,
                           const float* __restrict__ s_src,
                           const float* __restrict__ s_dst,
                           const float* __restrict__ denom,
                           const float* __restrict__ xp,
                           float* __restrict__ out) {
  const int gid = blockIdx.x * blockDim.x + threadIdx.x;
  if (gid >= N_EDGES * HEADS) return;
  const int e = gid >> 2, hd = gid & 3;
  const int s = src[e], d = dst[e];
  const float ex    = edge_logit_exp(s_src, s_dst, s, d, hd);
  const float alpha = ex / (denom[d * HEADS + hd] + GAT_EPS);
  const float* __restrict__ xs = xp  + (size_t)s * HID + hd * DHEAD;
  float*       __restrict__ od = out + (size_t)d * HID + hd * DHEAD;
#pragma unroll
  for (int i = 0; i < DHEAD; ++i)
    atomicAdd(&od[i], alpha * xs[i]);
}

__global__ void relu_inplace(float* __restrict__ p, int n) {
  const int gid = blockIdx.x * blockDim.x + threadIdx.x;
  if (gid < n) p[gid] = fmaxf(p[gid], 0.0f);
}

// ---------------------------------------------------------------------------
extern "C" void kernel_launch(void* const* d_in, const int* in_sizes, int n_in,
                              void* d_out, int out_size, void* d_ws, size_t ws_size,
                              hipStream_t stream) {
  const float* x        = (const float*)d_in[0];
  const float* w_embed  = (const float*)d_in[1];
  const float* b_embed  = (const float*)d_in[2];
  const float* lin_w0   = (const float*)d_in[3];
  const float* att_src0 = (const float*)d_in[4];
  const float* att_dst0 = (const float*)d_in[5];
  const float* lin_w1   = (const float*)d_in[6];
  const float* att_src1 = (const float*)d_in[7];
  const float* att_dst1 = (const float*)d_in[8];
  const int*   edge     = (const int*)d_in[9];
  const int*   src      = edge;            // edge_index[0]
  const int*   dst      = edge + N_EDGES;  // edge_index[1]
  float*       out      = (float*)d_out;

  // Workspace layout (floats): bufA/bufB reused across layers (~56 MB total)
  float* ws    = (float*)d_ws;
  float* bufA  = ws;                                   // h / layer-0 out  [N*64]
  float* bufB  = bufA + (size_t)N_NODES * HID;         // xp               [N*64]
  float* s_src = bufB + (size_t)N_NODES * HID;         // [N*4]
  float* s_dst = s_src + (size_t)N_NODES * HEADS;      // [N*4]
  float* denom = s_dst + (size_t)N_NODES * HEADS;      // [N*4]

  const int rowTiles = N_NODES / 16;                   // 6250, exact
  const dim3 gblk(128);
  const int nh   = N_NODES * HEADS;
  const int nhg  = (nh + 255) / 256;
  const int nfull = N_NODES * HID;
  const int nfg  = (nfull + 255) / 256;
  const int eh   = N_EDGES * HEADS;
  const int ehg  = (eh + 255) / 256;

  // node_embed: bufA = relu(x @ w_embed + b)
  gemm_wmma_f32<IN_DIM, true><<<rowTiles, gblk, 0, stream>>>(x, w_embed, b_embed, bufA);

  // ---- GAT layer 0 : input bufA, xp bufB, output overwrites bufA ----
  gemm_wmma_f32<HID, false><<<rowTiles, gblk, 0, stream>>>(bufA, lin_w0, nullptr, bufB);
  node_scores<<<nhg, 256, 0, stream>>>(bufB, att_src0, att_dst0, s_src, s_dst);
  zero_f32<<<nfg, 256, 0, stream>>>(bufA, nfull);      // bufA no longer needed as input
  zero_f32<<<nhg, 256, 0, stream>>>(denom, nh);
  edge_pass1<<<ehg, 256, 0, stream>>>(src, dst, s_src, s_dst, denom);
  edge_pass2<<<ehg, 256, 0, stream>>>(src, dst, s_src, s_dst, denom, bufB, bufA);

  // ---- GAT layer 1 : input bufA, xp bufB, output -> d_out ----
  gemm_wmma_f32<HID, false><<<rowTiles, gblk, 0, stream>>>(bufA, lin_w1, nullptr, bufB);
  node_scores<<<nhg, 256, 0, stream>>>(bufB, att_src1, att_dst1, s_src, s_dst);
  zero_f32<<<nfg, 256, 0, stream>>>(out, nfull);
  zero_f32<<<nhg, 256, 0, stream>>>(denom, nh);
  edge_pass1<<<ehg, 256, 0, stream>>>(src, dst, s_src, s_dst, denom);
  edge_pass2<<<ehg, 256, 0, stream>>>(src, dst, s_src, s_dst, denom, bufB, out);

  // gat_post ReLU
  relu_inplace<<<nfg, 256, 0, stream>>>(out, nfull);
}